// GNNModel_7533372637202
// MI455X (gfx1250) — compile-verified
//
#include <hip/hip_runtime.h>
#include <hip/hip_bf16.h>
#include <math.h>

#define N_NODES 100000
#define N_EDGES 1600000
#define F_DIM 128
#define H_DIM 128
#define C_DIM 16

typedef __attribute__((ext_vector_type(2))) float v2f;
typedef __attribute__((ext_vector_type(8))) float v8f;

// ---------------------------------------------------------------------------
// Degree / normalization
// ---------------------------------------------------------------------------
__global__ void init_deg_kernel(float* __restrict__ deg) {
  const int i = blockIdx.x * blockDim.x + threadIdx.x;
  if (i < N_NODES) deg[i] = 1.0f;  // self-loop weight
}

__global__ void deg_atomic_kernel(const int* __restrict__ dst,
                                  const float* __restrict__ ew,
                                  float* __restrict__ deg) {
  const int e = blockIdx.x * blockDim.x + threadIdx.x;
  if (e < N_EDGES) atomicAdd(&deg[dst[e]], ew[e]);
}

__global__ void dinv_kernel(float* __restrict__ deg) {
  const int i = blockIdx.x * blockDim.x + threadIdx.x;
  if (i < N_NODES) {
    const float d = deg[i];
    deg[i] = (d > 0.0f) ? rsqrtf(d) : 0.0f;  // in place: deg -> dinv
  }
}

// ---------------------------------------------------------------------------
// GEMM1: h = x @ W1   (fp32 WMMA 16x16x4)
// Block: 256 threads (8 waves). 64 rows per block; wave w owns col tile w.
// ---------------------------------------------------------------------------
__global__ __launch_bounds__(256) void gemm1_wmma(const float* __restrict__ x,
                                                  const float* __restrict__ W1,
                                                  float* __restrict__ h) {
  __shared__ float sX[64 * 132];  // padded stride 132 -> bank-conflict free A reads
  const int tid  = threadIdx.x;
  const int row0 = blockIdx.x * 64;

  // Stage 64x128 x-tile (zero-padded past N)
  for (int i = tid; i < 64 * 32; i += 256) {
    const int r = i >> 5, c = i & 31;
    float4 v = make_float4(0.f, 0.f, 0.f, 0.f);
    const int gr = row0 + r;
    if (gr < N_NODES) v = ((const float4*)x)[gr * 32 + c];
    *(float4*)&sX[r * 132 + c * 4] = v;
  }
  __syncthreads();

  const int wave = tid >> 5;
  const int lane = tid & 31;
  const int lrow = lane & 15;
  const int hi   = lane >> 4;           // 0: K={k0,k0+1}, 1: K={k0+2,k0+3}
  const int col  = wave * 16 + lrow;

  for (int t = 0; t < 4; ++t) {
    v8f acc = {};
    const float* ax = &sX[(t * 16 + lrow) * 132 + 2 * hi];
    for (int k0 = 0; k0 < 128; k0 += 4) {
      v2f a, b;
      a.x = ax[k0];
      a.y = ax[k0 + 1];
      const int kb = k0 + 2 * hi;
      b.x = W1[kb * 128 + col];          // W1 is 64KB, L0/L2 resident
      b.y = W1[(kb + 1) * 128 + col];
      acc = __builtin_amdgcn_wmma_f32_16x16x4_f32(false, a, false, b,
                                                  (short)0, acc, false, false);
    }
    const int rbase = row0 + t * 16 + 8 * hi;
#pragma unroll
    for (int r = 0; r < 8; ++r) {        // C/D: VGPR r -> M = r (+8 for hi lanes)
      const int gr = rbase + r;
      if (gr < N_NODES) h[gr * 128 + col] = acc[r];
    }
  }
}

// ---------------------------------------------------------------------------
// Layer-1 aggregation: self-loop init + weighted edge scatter (atomics)
// ---------------------------------------------------------------------------
__global__ void selfloop1_kernel(const float* __restrict__ h,
                                 const float* __restrict__ dinv,
                                 float* __restrict__ agg) {
  const int idx = blockIdx.x * blockDim.x + threadIdx.x;
  if (idx >= N_NODES * 128) return;
  const float di = dinv[idx >> 7];
  agg[idx] = h[idx] * di * di;           // norm(self) = dinv*1*dinv
}

__global__ void scatter1_kernel(const int* __restrict__ src,
                                const int* __restrict__ dst,
                                const float* __restrict__ ew,
                                const float* __restrict__ dinv,
                                const float* __restrict__ h,
                                float* __restrict__ agg) {
  const int gid  = blockIdx.x * blockDim.x + threadIdx.x;
  const int e    = gid >> 5;             // one edge per 32-lane group
  const int lane = threadIdx.x & 31;
  if (e >= N_EDGES) return;
  const int   s    = src[e];
  const int   d    = dst[e];
  const float norm = dinv[s] * ew[e] * dinv[d];
  const float4 v = ((const float4*)(h + s * 128))[lane];  // coalesced 512B gather
  float* o = agg + d * 128 + lane * 4;
  atomicAdd(o + 0, v.x * norm);
  atomicAdd(o + 1, v.y * norm);
  atomicAdd(o + 2, v.z * norm);
  atomicAdd(o + 3, v.w * norm);
}

// ---------------------------------------------------------------------------
// GEMM2: h2 = relu(agg1 + b1) @ W2  (bias+ReLU fused into A-fragment load)
// Block: 256 threads (8 waves) -> 128 rows; single 16-wide column tile.
// ---------------------------------------------------------------------------
__global__ __launch_bounds__(256) void gemm2_wmma(const float* __restrict__ agg1,
                                                  const float* __restrict__ b1,
                                                  const float* __restrict__ W2,
                                                  float* __restrict__ h2) {
  __shared__ float sW[128 * 16];
  __shared__ float sb[128];
  const int tid = threadIdx.x;
  for (int i = tid; i < 512; i += 256) ((float4*)sW)[i] = ((const float4*)W2)[i];
  if (tid < 128) sb[tid] = b1[tid];
  __syncthreads();

  const int wave = tid >> 5, lane = tid & 31;
  const int lrow = lane & 15, hi = lane >> 4;
  const int row0 = blockIdx.x * 128 + wave * 16;
  const int arow = row0 + lrow;

  v8f acc = {};
  for (int k0 = 0; k0 < 128; k0 += 4) {
    const int k = k0 + 2 * hi;
    float a0 = 0.f, a1 = 0.f;
    if (arow < N_NODES) {
      a0 = fmaxf(agg1[arow * 128 + k] + sb[k], 0.f);
      a1 = fmaxf(agg1[arow * 128 + k + 1] + sb[k + 1], 0.f);
    }
    v2f a, b;
    a.x = a0; a.y = a1;
    b.x = sW[k * 16 + lrow];
    b.y = sW[(k + 1) * 16 + lrow];
    acc = __builtin_amdgcn_wmma_f32_16x16x4_f32(false, a, false, b,
                                                (short)0, acc, false, false);
  }
  const int rbase = row0 + 8 * hi;
#pragma unroll
  for (int r = 0; r < 8; ++r) {
    const int gr = rbase + r;
    if (gr < N_NODES) h2[gr * 16 + lrow] = acc[r];
  }
}

// ---------------------------------------------------------------------------
// Layer-2 aggregation
// ---------------------------------------------------------------------------
__global__ void selfloop2_kernel(const float* __restrict__ h2,
                                 const float* __restrict__ dinv,
                                 float* __restrict__ agg) {
  const int idx = blockIdx.x * blockDim.x + threadIdx.x;
  if (idx >= N_NODES * 16) return;
  const float di = dinv[idx >> 4];
  agg[idx] = h2[idx] * di * di;
}

__global__ void scatter2_kernel(const int* __restrict__ src,
                                const int* __restrict__ dst,
                                const float* __restrict__ ew,
                                const float* __restrict__ dinv,
                                const float* __restrict__ h2,
                                float* __restrict__ agg) {
  const int gid = blockIdx.x * blockDim.x + threadIdx.x;
  const int e   = gid >> 2;              // one edge per 4-lane group
  const int sub = threadIdx.x & 3;
  if (e >= N_EDGES) return;
  const int   s    = src[e];
  const int   d    = dst[e];
  const float norm = dinv[s] * ew[e] * dinv[d];
  const float4 v = ((const float4*)(h2 + s * 16))[sub];
  float* o = agg + d * 16 + sub * 4;
  atomicAdd(o + 0, v.x * norm);
  atomicAdd(o + 1, v.y * norm);
  atomicAdd(o + 2, v.z * norm);
  atomicAdd(o + 3, v.w * norm);
}

// ---------------------------------------------------------------------------
// Softmax over C=16 (bias fused), one node per thread
// ---------------------------------------------------------------------------
__global__ void softmax_kernel(const float* __restrict__ agg2,
                               const float* __restrict__ b2,
                               float* __restrict__ out) {
  const int i = blockIdx.x * blockDim.x + threadIdx.x;
  if (i >= N_NODES) return;
  float v[16];
  const float4* p = (const float4*)(agg2 + (size_t)i * 16);
  float m = -3.402823466e+38f;
#pragma unroll
  for (int q = 0; q < 4; ++q) {
    const float4 t = p[q];
    v[4 * q + 0] = t.x + b2[4 * q + 0];
    v[4 * q + 1] = t.y + b2[4 * q + 1];
    v[4 * q + 2] = t.z + b2[4 * q + 2];
    v[4 * q + 3] = t.w + b2[4 * q + 3];
  }
#pragma unroll
  for (int c = 0; c < 16; ++c) m = fmaxf(m, v[c]);
  float s = 0.f;
#pragma unroll
  for (int c = 0; c < 16; ++c) { v[c] = expf(v[c] - m); s += v[c]; }
  const float inv = 1.0f / s;
  float4* o = (float4*)(out + (size_t)i * 16);
#pragma unroll
  for (int q = 0; q < 4; ++q)
    o[q] = make_float4(v[4 * q + 0] * inv, v[4 * q + 1] * inv,
                       v[4 * q + 2] * inv, v[4 * q + 3] * inv);
}

// ---------------------------------------------------------------------------
// Launcher
// ---------------------------------------------------------------------------
extern "C" void kernel_launch(void* const* d_in, const int* in_sizes, int n_in,
                              void* d_out, int out_size, void* d_ws, size_t ws_size,
                              hipStream_t stream) {
  const float* x  = (const float*)d_in[0];
  const int*   ei = (const int*)d_in[1];
  const float* ew = (const float*)d_in[2];
  const float* W1 = (const float*)d_in[3];
  const float* b1 = (const float*)d_in[4];
  const float* W2 = (const float*)d_in[5];
  const float* b2 = (const float*)d_in[6];
  float* out = (float*)d_out;

  const int* src = ei;
  const int* dst = ei + N_EDGES;

  // Workspace layout (floats). h region is reused for h2/agg2 after layer-1.
  float* ws   = (float*)d_ws;
  float* dinv = ws;                                       // N (deg -> dinv in place)
  float* h    = ws + N_NODES;                             // N*128
  float* agg1 = ws + N_NODES + (size_t)N_NODES * 128;     // N*128
  float* h2   = h;                                        // N*16 (alias, h is dead)
  float* agg2 = h + (size_t)N_NODES * 16;                 // N*16 (alias)

  const int T = 256;

  // Degree / symmetric normalization
  init_deg_kernel<<<(N_NODES + T - 1) / T, T, 0, stream>>>(dinv);
  deg_atomic_kernel<<<(N_EDGES + T - 1) / T, T, 0, stream>>>(dst, ew, dinv);
  dinv_kernel<<<(N_NODES + T - 1) / T, T, 0, stream>>>(dinv);

  // Layer 1
  gemm1_wmma<<<(N_NODES + 63) / 64, T, 0, stream>>>(x, W1, h);
  selfloop1_kernel<<<(N_NODES * 128 + T - 1) / T, T, 0, stream>>>(h, dinv, agg1);
  scatter1_kernel<<<(N_EDGES * 32 + T - 1) / T, T, 0, stream>>>(src, dst, ew, dinv, h, agg1);

  // Layer 2 (bias+ReLU fused into GEMM2 A-load)
  gemm2_wmma<<<(N_NODES + 127) / 128, T, 0, stream>>>(agg1, b1, W2, h2);
  selfloop2_kernel<<<(N_NODES * 16 + T - 1) / T, T, 0, stream>>>(h2, dinv, agg2);
  scatter2_kernel<<<(N_EDGES * 4 + T - 1) / T, T, 0, stream>>>(src, dst, ew, dinv, h2, agg2);

  // Softmax (+b2)
  softmax_kernel<<<(N_NODES + T - 1) / T, T, 0, stream>>>(agg2, b2, out);
}